// GCN_49649821941769
// MI455X (gfx1250) — compile-verified
//
#include <hip/hip_runtime.h>

typedef __attribute__((ext_vector_type(2))) float v2f;
typedef __attribute__((ext_vector_type(8))) float v8f;

// ---------------------------------------------------------------- utilities

__global__ void GCN_zero_kernel(float* __restrict__ p, long n) {
    long i = (long)blockIdx.x * blockDim.x + threadIdx.x;
    if (i < n) p[i] = 0.0f;
}

__global__ void GCN_degree_kernel(const long long* __restrict__ src,
                                  const long long* __restrict__ dst,
                                  float* __restrict__ deg_out,
                                  float* __restrict__ deg_in, int E) {
    int e = blockIdx.x * blockDim.x + threadIdx.x;
    if (e < E) {
        atomicAdd(&deg_out[(int)src[e]], 1.0f);
        atomicAdd(&deg_in[(int)dst[e]], 1.0f);
    }
}

__global__ void GCN_norm_kernel(float* __restrict__ p, int n) {
    int i = blockIdx.x * blockDim.x + threadIdx.x;
    if (i < n) p[i] = rsqrtf(fmaxf(p[i], 1.0f));
}

// ------------------------------------------------------------ WMMA GEMM
// Hout[N x DOUT] = A' @ W, where
//   FIRST:  A'[r][k] = Ain[r][k] * nsrc[r]
//   !FIRST: A'[r][k] = relu(Ain[r][k]*ndst[r] + bias[k]) * nsrc[r]
// Block = 128 threads (4 waves, wave32). Each block owns 16 rows; each wave
// owns DOUT/(16*4) column tiles. fp32 WMMA 16x16x4, k-loop over DIN.
template <int DIN, int DOUT, bool FIRST>
__global__ __launch_bounds__(128) void GCN_gemm_wmma(
    const float* __restrict__ Ain, const float* __restrict__ bias,
    const float* __restrict__ nsrc, const float* __restrict__ ndst,
    const float* __restrict__ W, float* __restrict__ Hout) {
    constexpr int TPW = DOUT / 16 / 4;  // column tiles per wave
    const int lane = threadIdx.x & 31;
    const int wave = threadIdx.x >> 5;
    const int half = lane >> 4;   // 0: lanes 0-15, 1: lanes 16-31
    const int l15  = lane & 15;
    const int row0 = blockIdx.x * 16;
    const int row  = row0 + l15;  // A-fragment row for this lane

    const float ns = nsrc[row];
    float nd = 0.0f;
    if (!FIRST) nd = ndst[row];

    v8f acc[TPW] = {};

    for (int k = 0; k < DIN; k += 4) {
        const int kb = k + 2 * half;  // this lane's K pair (even -> 8B aligned)
        // ---- A fragment (fused epilogue of previous layer)
        v2f a;
        {
            const float2 av = *(const float2*)(&Ain[(long)row * DIN + kb]);
            if (FIRST) {
                a.x = av.x * ns;
                a.y = av.y * ns;
            } else {
                const float2 bb = *(const float2*)(&bias[kb]);
                a.x = fmaxf(av.x * nd + bb.x, 0.0f) * ns;
                a.y = fmaxf(av.y * nd + bb.y, 0.0f) * ns;
            }
        }
        // ---- B fragments + WMMA
        for (int t = 0; t < TPW; ++t) {
            const int n0 = (wave * TPW + t) * 16;
            v2f b;
            b.x = W[(long)kb * DOUT + n0 + l15];
            b.y = W[(long)(kb + 1) * DOUT + n0 + l15];
            acc[t] = __builtin_amdgcn_wmma_f32_16x16x4_f32(
                false, a, false, b, (short)0, acc[t], false, false);
        }
    }
    // ---- store D (C/D layout: reg i -> M = i + 8*half, N = l15)
    for (int t = 0; t < TPW; ++t) {
        const int n0 = (wave * TPW + t) * 16;
        for (int i = 0; i < 8; ++i) {
            const int m = i + 8 * half;
            Hout[(long)(row0 + m) * DOUT + n0 + l15] = acc[t][i];
        }
    }
}

// ------------------------------------------------------------ edge scatter
// agg[dst[e]][:] += h[src[e]][:]  ; D/4 threads per edge, float4 gathers.
template <int D>
__global__ void GCN_scatter_kernel(const float* __restrict__ h,
                                   const long long* __restrict__ src,
                                   const long long* __restrict__ dst,
                                   float* __restrict__ agg, int E) {
    constexpr int TPE = D / 4;
    long gid = (long)blockIdx.x * blockDim.x + threadIdx.x;
    long e = gid / TPE;
    if (e >= E) return;
    const int f = (int)(gid % TPE) * 4;
    const int s = (int)src[e];
    const int d = (int)dst[e];
    const float4 v = *(const float4*)(&h[(long)s * D + f]);
    float* ap = &agg[(long)d * D + f];
    atomicAdd(ap + 0, v.x);
    atomicAdd(ap + 1, v.y);
    atomicAdd(ap + 2, v.z);
    atomicAdd(ap + 3, v.w);
}

// out = agg * ndst[row] + b2   (final layer, no relu), D = 64
__global__ void GCN_finalize_kernel(const float* __restrict__ agg,
                                    const float* __restrict__ ndst,
                                    const float* __restrict__ b,
                                    float* __restrict__ out, int n) {
    int i = blockIdx.x * blockDim.x + threadIdx.x;
    if (i < n) {
        const int r = i >> 6;
        const int f = i & 63;
        out[i] = agg[i] * ndst[r] + b[f];
    }
}

// ---------------------------------------------------------------- launcher

extern "C" void kernel_launch(void* const* d_in, const int* in_sizes, int n_in,
                              void* d_out, int out_size, void* d_ws, size_t ws_size,
                              hipStream_t stream) {
    const float*     feat = (const float*)d_in[0];
    const long long* ei   = (const long long*)d_in[1];  // int64 edge_index [2,E]
    const float*     W0   = (const float*)d_in[2];
    const float*     b0   = (const float*)d_in[3];
    const float*     W1   = (const float*)d_in[4];
    const float*     b1   = (const float*)d_in[5];
    const float*     W2   = (const float*)d_in[6];
    const float*     b2   = (const float*)d_in[7];
    float* out = (float*)d_out;

    const int N = in_sizes[0] / 128;   // 100000
    const int E = in_sizes[1] / 2;     // 1600000
    const long long* src = ei;
    const long long* dst = ei + E;

    // workspace layout (floats): nsrc | ndst | h[N*128] | aggA[N*128] | aggB[N*128]
    float* ws   = (float*)d_ws;
    float* nsrc = ws;
    float* ndst = ws + N;
    float* h    = ws + 2L * N;
    float* aggA = h + (long)N * 128;
    float* aggB = aggA + (long)N * 128;

    auto cdiv = [](long a, long b) { return (int)((a + b - 1) / b); };
    const int gemmGrid = N / 16;  // 6250, exact

    // degrees -> norms (nsrc/ndst double as degree accumulators)
    GCN_zero_kernel<<<cdiv(2L * N, 256), 256, 0, stream>>>(nsrc, 2L * N);
    GCN_degree_kernel<<<cdiv(E, 256), 256, 0, stream>>>(src, dst, nsrc, ndst, E);
    GCN_norm_kernel<<<cdiv(2L * N, 256), 256, 0, stream>>>(nsrc, 2 * N);

    // ---- layer 0: h = (feat*nsrc)@W0 ; aggA = scatter(h)
    GCN_zero_kernel<<<cdiv((long)N * 128, 256), 256, 0, stream>>>(aggA, (long)N * 128);
    GCN_gemm_wmma<128, 128, true><<<gemmGrid, 128, 0, stream>>>(
        feat, b0 /*unused*/, nsrc, ndst, W0, h);
    GCN_scatter_kernel<128><<<cdiv((long)E * 32, 256), 256, 0, stream>>>(
        h, src, dst, aggA, E);

    // ---- layer 1: h = relu(aggA*ndst+b0)*nsrc @ W1 ; aggB = scatter(h)
    GCN_zero_kernel<<<cdiv((long)N * 128, 256), 256, 0, stream>>>(aggB, (long)N * 128);
    GCN_gemm_wmma<128, 128, false><<<gemmGrid, 128, 0, stream>>>(
        aggA, b0, nsrc, ndst, W1, h);
    GCN_scatter_kernel<128><<<cdiv((long)E * 32, 256), 256, 0, stream>>>(
        h, src, dst, aggB, E);

    // ---- layer 2: h = relu(aggB*ndst+b1)*nsrc @ W2 ; aggA = scatter(h) ; out
    GCN_zero_kernel<<<cdiv((long)N * 64, 256), 256, 0, stream>>>(aggA, (long)N * 64);
    GCN_gemm_wmma<128, 64, false><<<gemmGrid, 128, 0, stream>>>(
        aggB, b1, nsrc, ndst, W2, h);
    GCN_scatter_kernel<64><<<cdiv((long)E * 16, 256), 256, 0, stream>>>(
        h, src, dst, aggA, E);

    GCN_finalize_kernel<<<cdiv((long)N * 64, 256), 256, 0, stream>>>(
        aggA, ndst, b2, out, N * 64);
}